// GCN1_56478819943013
// MI455X (gfx1250) — compile-verified
//
#include <hip/hip_runtime.h>

#define N_NODES  100000
#define N_EDGES  1600000
#define FEAT     32
#define HID      32
#define N_GRAPHS 64

typedef __attribute__((ext_vector_type(2))) float v2f;
typedef __attribute__((ext_vector_type(8))) float v8f;

// float atomic max via int/uint ordering trick (init memory to 0xFF800000 = -inf)
__device__ __forceinline__ void atomicMaxFloat(float* addr, float val) {
    if (val >= 0.0f) atomicMax((int*)addr, __float_as_int(val));
    else             atomicMin((unsigned int*)addr, __float_as_uint(val));
}

__global__ void fill_u32_kernel(unsigned int* __restrict__ p, unsigned int pattern, int n) {
    int i = blockIdx.x * blockDim.x + threadIdx.x;
    if (i < n) p[i] = pattern;
}

__global__ void degree_kernel(const long long* __restrict__ row, float* __restrict__ deg, int nE) {
    int e = blockIdx.x * blockDim.x + threadIdx.x;
    if (e < nE) unsafeAtomicAdd(&deg[(int)row[e]], 1.0f);
}

__global__ void dis_kernel(float* __restrict__ deg, int n) {
    int i = blockIdx.x * blockDim.x + threadIdx.x;
    if (i < n) { float d = deg[i]; deg[i] = (d > 0.0f) ? rsqrtf(d) : 0.0f; }
}

__global__ void norm_kernel(const long long* __restrict__ row, const long long* __restrict__ col,
                            const float* __restrict__ dis, float* __restrict__ norm, int nE) {
    int e = blockIdx.x * blockDim.x + threadIdx.x;
    if (e < nE) norm[e] = -dis[(int)row[e]] * dis[(int)col[e]];
}

// out[col[e]][:] += norm[e] * srcH[row[e]][:]  -- 8 lanes per edge, float4 gather + 4 f32 atomics
__global__ void prop_kernel(const float* __restrict__ srcH, float* __restrict__ dstH,
                            const float* __restrict__ norm,
                            const long long* __restrict__ row, const long long* __restrict__ col,
                            int nE) {
    int tid = blockIdx.x * blockDim.x + threadIdx.x;
    int e = tid >> 3;
    if (e >= nE) return;
    int part = (tid & 7) << 2;
    float w = norm[e];
    int r = (int)row[e];
    int c = (int)col[e];
    float4 v = *(const float4*)(srcH + r * FEAT + part);
    float* dst = dstH + c * FEAT + part;
    unsafeAtomicAdd(dst + 0, w * v.x);
    unsafeAtomicAdd(dst + 1, w * v.y);
    unsafeAtomicAdd(dst + 2, w * v.z);
    unsafeAtomicAdd(dst + 3, w * v.w);
}

// h = Tx0@Wc0 + Tx1@Wc1 + (2*P2 - Tx0)@Wc2 + b, fused with per-graph max-pool into g[64,32].
// One wave per 16-node tile. K = 96 (3 orders x 32), 24 k-steps of 4, 2 n-tiles of 16.
__global__ void cheb_pool_wmma_kernel(const float* __restrict__ x,
                                      const float* __restrict__ p1,
                                      const float* __restrict__ p2,
                                      const float* __restrict__ Wc,   // [3,32,32]
                                      const float* __restrict__ bc,   // [32]
                                      const long long* __restrict__ batch,
                                      float* __restrict__ g,          // [64,32] (-inf initialized)
                                      int nTiles) {
    int wave = (blockIdx.x * blockDim.x + threadIdx.x) >> 5;
    int lane = threadIdx.x & 31;
    if (wave >= nTiles) return;               // uniform per wave -> EXEC stays all-1 for WMMA

    int mbase = wave * 16;
    int mrow  = mbase + (lane & 15);          // A: lanes 0-15 hold M, lanes 16-31 same M, K+2
    int koff  = (lane >> 4) << 1;             // 0 or 2
    int ncol  = lane & 15;                    // B/C/D column within n-tile

    v8f acc0 = {0.f,0.f,0.f,0.f,0.f,0.f,0.f,0.f};
    v8f acc1 = {0.f,0.f,0.f,0.f,0.f,0.f,0.f,0.f};

#pragma unroll
    for (int kb = 0; kb < 24; ++kb) {
        int kg = kb * 4 + koff;               // global K in 0..95
        int order = kg >> 5;
        int kl = kg & 31;

        const float* src = (order == 0) ? x : ((order == 1) ? p1 : p2);
        float a0 = src[mrow * FEAT + kl];
        float a1 = src[mrow * FEAT + kl + 1];
        if (order == 2) {                     // Tx2 = 2*prop(Tx1) - Tx0, fused into A load
            a0 = 2.0f * a0 - x[mrow * FEAT + kl];
            a1 = 2.0f * a1 - x[mrow * FEAT + kl + 1];
        }
        v2f a; a.x = a0; a.y = a1;

        const float* wrow = Wc + order * (32 * 32) + kl * 32;   // W[order][kl][*]
        v2f b0, b1;
        b0.x = wrow[ncol];        b0.y = wrow[32 + ncol];        // n-tile 0 (cols 0-15)
        b1.x = wrow[16 + ncol];   b1.y = wrow[32 + 16 + ncol];   // n-tile 1 (cols 16-31)

        acc0 = __builtin_amdgcn_wmma_f32_16x16x4_f32(false, a, false, b0, (short)0, acc0, false, false);
        acc1 = __builtin_amdgcn_wmma_f32_16x16x4_f32(false, a, false, b1, (short)0, acc1, false, false);
    }

    // D layout: VGPR r -> row mbase+r (lanes 0-15) / mbase+8+r (lanes 16-31), col = lane&15
    int rbase = mbase + ((lane >> 4) << 3);
    float bias0 = bc[ncol];
    float bias1 = bc[ncol + 16];
#pragma unroll
    for (int r = 0; r < 8; ++r) {
        int node = rbase + r;
        int gidx = (int)batch[node] * HID;
        atomicMaxFloat(&g[gidx + ncol],      acc0[r] + bias0);
        atomicMaxFloat(&g[gidx + ncol + 16], acc1[r] + bias1);
    }
}

// Generic row-major C[M,N] = act(A[M,K] @ B[K,N] + bias); one wave per 16x16 tile.
__global__ void gemm_relu_wmma_kernel(const float* __restrict__ A, const float* __restrict__ B,
                                      const float* __restrict__ bias, float* __restrict__ C,
                                      int M, int K, int Nn, int relu, int nTiles, int ntCols) {
    int wave = (blockIdx.x * blockDim.x + threadIdx.x) >> 5;
    int lane = threadIdx.x & 31;
    if (wave >= nTiles) return;

    int mt = wave / ntCols;
    int nt = wave % ntCols;
    int mbase = mt * 16;
    int nbase = nt * 16;
    int mrow = mbase + (lane & 15);
    int koff = (lane >> 4) << 1;
    int ncol = nbase + (lane & 15);

    v8f acc = {0.f,0.f,0.f,0.f,0.f,0.f,0.f,0.f};
    int ksteps = K >> 2;
    for (int kb = 0; kb < ksteps; ++kb) {
        int kg = kb * 4 + koff;
        v2f a, b;
        a.x = A[mrow * K + kg];
        a.y = A[mrow * K + kg + 1];
        b.x = B[kg * Nn + ncol];
        b.y = B[(kg + 1) * Nn + ncol];
        acc = __builtin_amdgcn_wmma_f32_16x16x4_f32(false, a, false, b, (short)0, acc, false, false);
    }

    int rbase = mbase + ((lane >> 4) << 3);
    float bv = bias[ncol];
#pragma unroll
    for (int r = 0; r < 8; ++r) {
        float v = acc[r] + bv;
        if (relu) v = fmaxf(v, 0.0f);
        C[(rbase + r) * Nn + ncol] = v;
    }
}

// out[64,4] = h2[64,512] @ W3[512,4] + b3
__global__ void final_layer_kernel(const float* __restrict__ h2, const float* __restrict__ W3,
                                   const float* __restrict__ b3, float* __restrict__ out) {
    int tid = blockIdx.x * blockDim.x + threadIdx.x;
    if (tid >= N_GRAPHS * 4) return;
    int m = tid >> 2;
    int n = tid & 3;
    float s = b3[n];
    const float* hrow = h2 + m * 512;
    for (int k = 0; k < 512; ++k) s += hrow[k] * W3[k * 4 + n];
    out[tid] = s;
}

extern "C" void kernel_launch(void* const* d_in, const int* in_sizes, int n_in,
                              void* d_out, int out_size, void* d_ws, size_t ws_size,
                              hipStream_t stream) {
    (void)in_sizes; (void)n_in; (void)out_size; (void)ws_size;
    const float*     x     = (const float*)d_in[0];
    const long long* eidx  = (const long long*)d_in[1];  // int64 [2, E]
    const long long* batch = (const long long*)d_in[2];  // int64 [N]
    const float*     Wc    = (const float*)d_in[3];      // [3,32,32]
    const float*     bc    = (const float*)d_in[4];      // [32]
    const float*     W1    = (const float*)d_in[5];      // [32,1024]
    const float*     b1    = (const float*)d_in[6];
    const float*     W2    = (const float*)d_in[7];      // [1024,512]
    const float*     b2    = (const float*)d_in[8];
    const float*     W3    = (const float*)d_in[9];      // [512,4]
    const float*     b3    = (const float*)d_in[10];
    float* out = (float*)d_out;

    const long long* row = eidx;
    const long long* col = eidx + N_EDGES;

    // workspace layout (floats)
    float* ws   = (float*)d_ws;
    float* deg  = ws;                                   // [N]       (becomes dis in place)
    float* p1   = deg + N_NODES;                        // [N,32]    prop(x)
    float* p2   = p1 + N_NODES * FEAT;                  // [N,32]    prop(p1)
    float* nrm  = p2 + N_NODES * FEAT;                  // [E]
    float* g    = nrm + N_EDGES;                        // [64,32]
    float* h1   = g + N_GRAPHS * HID;                   // [64,1024]
    float* h2   = h1 + N_GRAPHS * 1024;                 // [64,512]

    const int BLK = 256;

    // 1. init: zero deg+p1+p2 (contiguous), -inf pattern for g
    {
        int nz = N_NODES * (1 + 2 * FEAT);
        fill_u32_kernel<<<(nz + BLK - 1) / BLK, BLK, 0, stream>>>((unsigned int*)deg, 0u, nz);
        int ng = N_GRAPHS * HID;
        fill_u32_kernel<<<(ng + BLK - 1) / BLK, BLK, 0, stream>>>((unsigned int*)g, 0xFF800000u, ng);
    }

    // 2. degree -> rsqrt -> edge norms
    degree_kernel<<<(N_EDGES + BLK - 1) / BLK, BLK, 0, stream>>>(row, deg, N_EDGES);
    dis_kernel<<<(N_NODES + BLK - 1) / BLK, BLK, 0, stream>>>(deg, N_NODES);
    norm_kernel<<<(N_EDGES + BLK - 1) / BLK, BLK, 0, stream>>>(row, col, deg, nrm, N_EDGES);

    // 3. two propagation passes (Tx1 = L x ; P2 = L Tx1)
    {
        long long nt = (long long)N_EDGES * 8;
        int blocks = (int)((nt + BLK - 1) / BLK);
        prop_kernel<<<blocks, BLK, 0, stream>>>(x, p1, nrm, row, col, N_EDGES);
        prop_kernel<<<blocks, BLK, 0, stream>>>(p1, p2, nrm, row, col, N_EDGES);
    }

    // 4. fused Chebyshev combine (FP32 WMMA) + global max pool into g
    {
        int tiles = N_NODES / 16;                       // 6250, exact
        int waves_per_block = BLK / 32;
        int blocks = (tiles + waves_per_block - 1) / waves_per_block;
        cheb_pool_wmma_kernel<<<blocks, BLK, 0, stream>>>(x, p1, p2, Wc, bc, batch, g, tiles);
    }

    // 5. MLP head: two WMMA GEMM+ReLU layers, then tiny final layer
    {
        int tiles1 = (N_GRAPHS / 16) * (1024 / 16);     // 256
        gemm_relu_wmma_kernel<<<(tiles1 * 32 + BLK - 1) / BLK, BLK, 0, stream>>>(
            g, W1, b1, h1, N_GRAPHS, 32, 1024, 1, tiles1, 1024 / 16);

        int tiles2 = (N_GRAPHS / 16) * (512 / 16);      // 128
        gemm_relu_wmma_kernel<<<(tiles2 * 32 + BLK - 1) / BLK, BLK, 0, stream>>>(
            h1, W2, b2, h2, N_GRAPHS, 1024, 512, 1, tiles2, 512 / 16);

        final_layer_kernel<<<1, 256, 0, stream>>>(h2, W3, b3, out);
    }
}